// ContrastiveLoss_82300163326281
// MI455X (gfx1250) — compile-verified
//
#include <hip/hip_runtime.h>

// ContrastiveLoss for MI455X (gfx1250): f32 WMMA gram matrix, two-pass
// (stats pass + weighted-logsumexp pass), deterministic reductions.
// k_gemm: A fragments resident in VGPRs, double-buffered LDS B tiles filled
// with GLOBAL_LOAD_ASYNC_TO_LDS_B128 (no VGPR staging, ASYNCcnt-tracked,
// overlapped with the 64-WMMA block), chunked B fragment preload from LDS,
// dual accumulator chains, static buffer indices (2x unroll).

typedef __attribute__((ext_vector_type(2))) float v2f;
typedef __attribute__((ext_vector_type(8))) float v8f;

#define BN      8192
#define DIM     256
#define NPOS    (BN * 8)
#define NSPLIT  8
#define MBLK    128          // 8 waves x 16-row M tiles per block
#define NT_PER_SPLIT ((BN / 16) / NSPLIT)   // 64 N tiles per split (even)
#define LDSB_STRIDE 260      // padded row stride (floats); 260*4B = 65*16B -> f4 aligned, conflict-free
#define EPSF    1e-8f

// ---- order-preserving float<->uint encoding for atomic min/max ----
__device__ __forceinline__ unsigned fenc(float x) {
    unsigned u = __float_as_uint(x);
    return (u & 0x80000000u) ? ~u : (u | 0x80000000u);
}
__device__ __forceinline__ float fdec(unsigned u) {
    return __uint_as_float((u & 0x80000000u) ? (u & 0x7fffffffu) : ~u);
}

// scal slots (4B words): 0 inv_temp(f32), 1 vmin_u, 2 vmax_u, 3 negmin(f32), 4 inv_nrange(f32)

__global__ void k_init(const float* __restrict__ temperature, float* __restrict__ scal) {
    if (threadIdx.x == 0) {
        float t = temperature[0];
        float sp = (t > 20.0f) ? t : log1pf(__expf(t));   // softplus
        scal[0] = 1.0f / sp;
        ((unsigned*)scal)[1] = 0xFFFFFFFFu;  // vmin slot init (encoded +max)
        ((unsigned*)scal)[2] = 0u;           // vmax slot init (encoded -max)
    }
}

__global__ void k_normalize(const float* __restrict__ emb, float* __restrict__ z,
                            unsigned* __restrict__ rowmax_u,
                            unsigned* __restrict__ negmin_u,
                            unsigned* __restrict__ negmax_u) {
    const int lane = threadIdx.x & 31;
    const int wave = (blockIdx.x * blockDim.x + threadIdx.x) >> 5;
    const int nwave = (gridDim.x * blockDim.x) >> 5;
    for (int r = wave; r < BN; r += nwave) {
        const float4* src = (const float4*)(emb + (size_t)r * DIM);
        float4 a = src[lane * 2 + 0];
        float4 b = src[lane * 2 + 1];
        float ss = a.x*a.x + a.y*a.y + a.z*a.z + a.w*a.w
                 + b.x*b.x + b.y*b.y + b.z*b.z + b.w*b.w;
        #pragma unroll
        for (int m = 1; m < 32; m <<= 1) ss += __shfl_xor(ss, m);
        float scale = 1.0f / fmaxf(sqrtf(ss), 1e-12f);
        float4* dst = (float4*)(z + (size_t)r * DIM);
        a.x *= scale; a.y *= scale; a.z *= scale; a.w *= scale;
        b.x *= scale; b.y *= scale; b.z *= scale; b.w *= scale;
        dst[lane * 2 + 0] = a;
        dst[lane * 2 + 1] = b;
        if (lane == 0) {
            rowmax_u[r] = 0u;           // encoded -max
            negmin_u[r] = 0xFFFFFFFFu;  // encoded +max
            negmax_u[r] = 0u;
        }
    }
}

__global__ void k_posminmax(const float* __restrict__ pos_vals, float* __restrict__ scal) {
    __shared__ unsigned smin[256], smax[256];
    unsigned lmin = 0xFFFFFFFFu, lmax = 0u;
    for (int i = blockIdx.x * blockDim.x + threadIdx.x; i < NPOS; i += gridDim.x * blockDim.x) {
        unsigned e = fenc(pos_vals[i]);
        lmin = min(lmin, e);
        lmax = max(lmax, e);
    }
    smin[threadIdx.x] = lmin; smax[threadIdx.x] = lmax;
    __syncthreads();
    for (int off = 128; off > 0; off >>= 1) {
        if (threadIdx.x < off) {
            smin[threadIdx.x] = min(smin[threadIdx.x], smin[threadIdx.x + off]);
            smax[threadIdx.x] = max(smax[threadIdx.x], smax[threadIdx.x + off]);
        }
        __syncthreads();
    }
    if (threadIdx.x == 0) {
        atomicMin(((unsigned*)scal) + 1, smin[0]);
        atomicMax(((unsigned*)scal) + 2, smax[0]);
    }
}

// PASS 1: per-row max over all cols + per-row neg min/max (encoded atomics).
// PASS 2: per-row weighted exp-sum partials (unique writer per (split,row)).
template <int PASS>
__global__ __launch_bounds__(256, 1) void k_gemm(
        const float* __restrict__ z, const float* __restrict__ scal,
        unsigned* __restrict__ rowmax_u, unsigned* __restrict__ negmin_u,
        unsigned* __restrict__ negmax_u, const float* __restrict__ rowmaxf,
        float* __restrict__ rowS) {
    __shared__ float ldsB[2][16 * LDSB_STRIDE];

    const int wave  = threadIdx.x >> 5;
    const int lane  = threadIdx.x & 31;
    const int mtile = blockIdx.x * MBLK + wave * 16;
    const int arow  = mtile + (lane & 15);
    const int kb    = (lane >> 4) * 2;          // K sub-offset for A/B fragments
    const float invtemp = scal[0];

    // per-thread static byte offsets for the cooperative async tile copy
    // (4x 16B per thread covers the 16x256-f32 tile)
    unsigned g_stat[4], l_stat[4];
    #pragma unroll
    for (int i = 0; i < 4; ++i) {
        int f4 = threadIdx.x + i * 256;         // flat float4 index 0..1023
        int rr = f4 >> 6, c4 = f4 & 63;
        g_stat[i] = (unsigned)(rr * (DIM * 4) + c4 * 16);
        l_stat[i] = (unsigned)(rr * (LDSB_STRIDE * 4) + c4 * 16);
    }
    // LDS byte offsets of the two buffers (flat->LDS keeps offset in low 32 bits)
    unsigned ldsbase[2];
    ldsbase[0] = (unsigned)(unsigned long long)&ldsB[0][0];
    ldsbase[1] = (unsigned)(unsigned long long)&ldsB[1][0];

    // async global->LDS copy of one 16-row B tile (no VGPR data staging)
    auto async_copy_tile = [&](int buf, int nt) {
        #pragma unroll
        for (int i = 0; i < 4; ++i) {
            unsigned ldsoff = ldsbase[buf] + l_stat[i];
            unsigned goff   = (unsigned)(nt * (16 * DIM * 4)) + g_stat[i];
            asm volatile("global_load_async_to_lds_b128 %0, %1, %2"
                         :: "v"(ldsoff), "v"(goff), "s"(z)
                         : "memory");
        }
    };

    // A fragments for this wave's 16 rows, all of K=256: 64 x v2f = 128 VGPRs.
    v2f a[64];
    #pragma unroll
    for (int k = 0; k < 64; ++k)
        a[k] = *(const v2f*)(z + (size_t)arow * DIM + k * 4 + kb);

    float amax[8], nmin[8], nmax[8], ssum[8], rmax8[8];
    #pragma unroll
    for (int v = 0; v < 8; ++v) {
        amax[v] = -__builtin_inff();
        nmin[v] =  __builtin_inff();
        nmax[v] = -__builtin_inff();
        ssum[v] = 0.0f;
        rmax8[v] = 0.0f;
    }
    float negmin = 0.0f, invnr = 0.0f;
    if (PASS == 2) {
        negmin = scal[3];
        invnr  = scal[4];
        #pragma unroll
        for (int v = 0; v < 8; ++v)
            rmax8[v] = rowmaxf[mtile + v + ((lane >> 4) << 3)];
    }

    const int nt0 = blockIdx.y * NT_PER_SPLIT;

    // one tile step: async-copy tile preNt -> ldsB[buf^1] (overlapped),
    // 64 WMMAs from ldsB[buf], per-element stats, wait async, barrier.
    auto tile_body = [&](int buf, int nt, bool doPre, int preNt) {
        if (doPre) async_copy_tile(buf ^ 1, preNt);

        v8f acc0 = {0.f, 0.f, 0.f, 0.f, 0.f, 0.f, 0.f, 0.f};
        v8f acc1 = {0.f, 0.f, 0.f, 0.f, 0.f, 0.f, 0.f, 0.f};
        const float* bbase = &ldsB[buf][(lane & 15) * LDSB_STRIDE + kb];
        #pragma unroll
        for (int kc = 0; kc < 4; ++kc) {
            v2f b[16];
            #pragma unroll
            for (int k = 0; k < 16; ++k)
                b[k] = *(const v2f*)(bbase + (kc * 16 + k) * 4);
            #pragma unroll
            for (int k = 0; k < 16; k += 2) {
                acc0 = __builtin_amdgcn_wmma_f32_16x16x4_f32(
                           false, a[kc * 16 + k],     false, b[k],     (short)0, acc0, false, false);
                acc1 = __builtin_amdgcn_wmma_f32_16x16x4_f32(
                           false, a[kc * 16 + k + 1], false, b[k + 1], (short)0, acc1, false, false);
            }
        }

        const int col = nt * 16 + (lane & 15);
        #pragma unroll
        for (int v = 0; v < 8; ++v) {
            const int row = mtile + v + ((lane >> 4) << 3);
            const float val = (acc0[v] + acc1[v]) * invtemp;
            const int d = (col - row) & (BN - 1);   // 0=diag, 1..8=pos, >8=neg
            if (PASS == 1) {
                amax[v] = fmaxf(amax[v], val);
                if (d > 8) {
                    nmin[v] = fminf(nmin[v], val);
                    nmax[v] = fmaxf(nmax[v], val);
                }
            } else {
                const float sh = val - rmax8[v];
                const float w  = (d > 8) ? ((sh - negmin) * invnr + 1.0f) : 1.0f;
                ssum[v] += w * __expf(sh);
            }
        }

        asm volatile("s_wait_asynccnt 0x0" ::: "memory");
        __syncthreads();
    };

    // ---- prologue: async-copy tile nt0 -> ldsB[0] ----
    async_copy_tile(0, nt0);
    asm volatile("s_wait_asynccnt 0x0" ::: "memory");
    __syncthreads();

    // ---- main loop: 2x unrolled so LDS buffer indices are static ----
    for (int it = 0; it < NT_PER_SPLIT; it += 2) {
        const int nt = nt0 + it;
        tile_body(0, nt,     true,                       nt + 1);
        tile_body(1, nt + 1, (it + 2 < NT_PER_SPLIT),    nt + 2);
    }

    // reduce across the 16 lanes within each half (rows differ between halves)
    #pragma unroll
    for (int v = 0; v < 8; ++v) {
        #pragma unroll
        for (int m = 1; m < 16; m <<= 1) {
            if (PASS == 1) {
                amax[v] = fmaxf(amax[v], __shfl_xor(amax[v], m));
                nmin[v] = fminf(nmin[v], __shfl_xor(nmin[v], m));
                nmax[v] = fmaxf(nmax[v], __shfl_xor(nmax[v], m));
            } else {
                ssum[v] += __shfl_xor(ssum[v], m);
            }
        }
    }
    if ((lane & 15) == 0) {
        #pragma unroll
        for (int v = 0; v < 8; ++v) {
            const int row = mtile + v + ((lane >> 4) << 3);
            if (PASS == 1) {
                atomicMax(&rowmax_u[row], fenc(amax[v]));
                atomicMin(&negmin_u[row], fenc(nmin[v]));
                atomicMax(&negmax_u[row], fenc(nmax[v]));
            } else {
                rowS[blockIdx.y * BN + row] = ssum[v];
            }
        }
    }
}

__global__ void k_globalneg(const unsigned* __restrict__ rowmax_u,
                            const unsigned* __restrict__ negmin_u,
                            const unsigned* __restrict__ negmax_u,
                            float* __restrict__ rowmaxf, float* __restrict__ scal) {
    __shared__ float smn[256], smx[256];
    float gmin = __builtin_inff(), gmax = -__builtin_inff();
    for (int i = threadIdx.x; i < BN; i += 256) {
        float m = fdec(rowmax_u[i]);
        rowmaxf[i] = m;
        gmin = fminf(gmin, fdec(negmin_u[i]) - m);
        gmax = fmaxf(gmax, fdec(negmax_u[i]) - m);
    }
    smn[threadIdx.x] = gmin; smx[threadIdx.x] = gmax;
    __syncthreads();
    for (int off = 128; off > 0; off >>= 1) {
        if (threadIdx.x < off) {
            smn[threadIdx.x] = fminf(smn[threadIdx.x], smn[threadIdx.x + off]);
            smx[threadIdx.x] = fmaxf(smx[threadIdx.x], smx[threadIdx.x + off]);
        }
        __syncthreads();
    }
    if (threadIdx.x == 0) {
        scal[3] = smn[0];
        scal[4] = 1.0f / (smx[0] - smn[0] + EPSF);
    }
}

__global__ void k_rowlse(const float* __restrict__ rowS, float* __restrict__ rowlse) {
    int r = blockIdx.x * blockDim.x + threadIdx.x;
    if (r < BN) {
        float s = 0.0f;
        #pragma unroll
        for (int p = 0; p < NSPLIT; ++p) s += rowS[p * BN + r];  // fixed order
        rowlse[r] = logf(s);
    }
}

__global__ void k_pairs(const float* __restrict__ z,
                        const int* __restrict__ pos_row, const int* __restrict__ pos_col,
                        const float* __restrict__ pos_vals,
                        const float* __restrict__ rowmaxf, const float* __restrict__ rowlse,
                        const float* __restrict__ scal, float* __restrict__ pairv) {
    const int lane = threadIdx.x & 31;
    const int wave = (blockIdx.x * blockDim.x + threadIdx.x) >> 5;
    const int nwave = (gridDim.x * blockDim.x) >> 5;
    const float invtemp = scal[0];
    const float vmin = fdec(((const unsigned*)scal)[1]);
    const float vmax = fdec(((const unsigned*)scal)[2]);
    const float invp = 1.0f / (vmax - vmin + EPSF);   // pos_w = (vmax - v) * invp
    for (int p = wave; p < NPOS; p += nwave) {
        const int r = pos_row[p], c = pos_col[p];
        const float4* zr = (const float4*)(z + (size_t)r * DIM);
        const float4* zc = (const float4*)(z + (size_t)c * DIM);
        float4 a0 = zr[lane * 2], a1 = zr[lane * 2 + 1];
        float4 b0 = zc[lane * 2], b1 = zc[lane * 2 + 1];
        float d = a0.x*b0.x + a0.y*b0.y + a0.z*b0.z + a0.w*b0.w
                + a1.x*b1.x + a1.y*b1.y + a1.z*b1.z + a1.w*b1.w;
        #pragma unroll
        for (int m = 1; m < 32; m <<= 1) d += __shfl_xor(d, m);
        if (lane == 0) {
            float sh = d * invtemp - rowmaxf[r];
            float lp = sh - rowlse[r];
            float pw = (vmax - pos_vals[p]) * invp;
            pairv[p] = lp * pw;
        }
    }
}

__global__ void k_final(const float* __restrict__ pairv, float* __restrict__ out) {
    __shared__ float s[256];
    float acc = 0.0f;
    for (int i = threadIdx.x; i < NPOS; i += 256) acc += pairv[i];  // fixed order
    s[threadIdx.x] = acc;
    __syncthreads();
    for (int off = 128; off > 0; off >>= 1) {
        if (threadIdx.x < off) s[threadIdx.x] += s[threadIdx.x + off];
        __syncthreads();
    }
    if (threadIdx.x == 0) out[0] = -s[0] / (float)NPOS;
}

extern "C" void kernel_launch(void* const* d_in, const int* in_sizes, int n_in,
                              void* d_out, int out_size, void* d_ws, size_t ws_size,
                              hipStream_t stream) {
    const float* emb      = (const float*)d_in[0];
    const float* pos_vals = (const float*)d_in[1];
    const float* temper   = (const float*)d_in[2];
    const int*   pos_row  = (const int*)d_in[3];
    const int*   pos_col  = (const int*)d_in[4];
    float* out = (float*)d_out;

    // ---- workspace layout ----
    float*    z        = (float*)d_ws;                 // BN*DIM
    unsigned* rowmax_u = (unsigned*)(z + (size_t)BN * DIM);
    unsigned* negmin_u = rowmax_u + BN;
    unsigned* negmax_u = negmin_u + BN;
    float*    rowmaxf  = (float*)(negmax_u + BN);
    float*    rowlse   = rowmaxf + BN;
    float*    rowS     = rowlse + BN;                  // NSPLIT*BN
    float*    pairv    = rowS + (size_t)NSPLIT * BN;   // NPOS
    float*    scal     = pairv + NPOS;                 // 16 words

    k_init<<<1, 32, 0, stream>>>(temper, scal);
    k_normalize<<<256, 256, 0, stream>>>(emb, z, rowmax_u, negmin_u, negmax_u);
    k_posminmax<<<64, 256, 0, stream>>>(pos_vals, scal);
    k_gemm<1><<<dim3(BN / MBLK, NSPLIT), 256, 0, stream>>>(
        z, scal, rowmax_u, negmin_u, negmax_u, rowmaxf, rowS);
    k_globalneg<<<1, 256, 0, stream>>>(rowmax_u, negmin_u, negmax_u, rowmaxf, scal);
    k_gemm<2><<<dim3(BN / MBLK, NSPLIT), 256, 0, stream>>>(
        z, scal, rowmax_u, negmin_u, negmax_u, rowmaxf, rowS);
    k_rowlse<<<32, 256, 0, stream>>>(rowS, rowlse);
    k_pairs<<<512, 256, 0, stream>>>(z, pos_row, pos_col, pos_vals, rowmaxf, rowlse, scal, pairv);
    k_final<<<1, 256, 0, stream>>>(pairv, out);
}